// GNNmodule_1769526526172
// MI455X (gfx1250) — compile-verified
//
#include <hip/hip_runtime.h>
#include <hip/hip_bf16.h>

typedef __attribute__((ext_vector_type(2))) float v2f;
typedef __attribute__((ext_vector_type(4))) float v4f;
typedef __attribute__((ext_vector_type(8))) float v8f;

#define C_CH   32
#define BN_EPS 1e-5f

// ---------------------------------------------------------------------------
// K0: init deg with self-loop fill (1.0, IMPROVED=False) and zero the BN stats
// ---------------------------------------------------------------------------
__global__ void k_init(float* __restrict__ deg, float* __restrict__ stats, int n) {
    int i = blockIdx.x * blockDim.x + threadIdx.x;
    if (i < n)   deg[i]   = 1.0f;
    if (i < 128) stats[i] = 0.0f;
}

// ---------------------------------------------------------------------------
// K1: deg[dst] += ew  (weighted in-degree), relaxed agent-scope f32 atomics
// ---------------------------------------------------------------------------
__global__ void k_deg(const int* __restrict__ dst, const float* __restrict__ ew,
                      float* __restrict__ deg, int e) {
    int i = blockIdx.x * blockDim.x + threadIdx.x;
    if (i < e)
        __hip_atomic_fetch_add(&deg[dst[i]], ew[i],
                               __ATOMIC_RELAXED, __HIP_MEMORY_SCOPE_AGENT);
}

// ---------------------------------------------------------------------------
// K2: dinv = deg > 0 ? rsqrt(deg) : 0   (in place)
// ---------------------------------------------------------------------------
__global__ void k_dinv(float* __restrict__ deg, int n) {
    int i = blockIdx.x * blockDim.x + threadIdx.x;
    if (i < n) {
        float d = deg[i];
        deg[i] = (d > 0.0f) ? rsqrtf(d) : 0.0f;
    }
}

// ---------------------------------------------------------------------------
// K3: xw = x @ W via V_WMMA_F32_16X16X4_F32.
// One wave = one 16-row tile of x (16x32) times W (32x32) -> 16x32 out slab.
// Two v8f accumulators (N-halves 0..15 / 16..31), K=32 in 8 steps of 4.
//
// A layout (16x4 f32): lanes 0-15 hold M=lane, VGPR0=K+0, VGPR1=K+1;
//                      lanes 16-31 hold M=lane-16, VGPR0=K+2, VGPR1=K+3.
// B layout mirrored with N in place of M.
// C/D layout: VGPR r -> M = r + (lane>=16 ? 8 : 0), N = lane&15.
// ---------------------------------------------------------------------------
__global__ void k_gemm_wmma(const float* __restrict__ x, const float* __restrict__ W,
                            float* __restrict__ xw, int n) {
    const int lane = threadIdx.x & 31;
    const int wave = threadIdx.x >> 5;
    const int tile = blockIdx.x * (blockDim.x >> 5) + wave;
    const int ntiles = (n + 15) >> 4;
    if (tile >= ntiles) return;               // wave-uniform: EXEC stays all-1s

    const int half = lane >> 4;               // 0 | 1
    const int lid  = lane & 15;
    int arow = tile * 16 + lid;
    if (arow >= n) arow = n - 1;              // clamp loads on (non-existent) tail

    v8f c0 = {};
    v8f c1 = {};
#pragma unroll
    for (int kk = 0; kk < C_CH; kk += 4) {
        const int kb = kk + half * 2;
        v2f a, b0, b1;
        a.x  = x[arow * C_CH + kb];
        a.y  = x[arow * C_CH + kb + 1];
        b0.x = W[kb * C_CH + lid];
        b0.y = W[(kb + 1) * C_CH + lid];
        b1.x = W[kb * C_CH + lid + 16];
        b1.y = W[(kb + 1) * C_CH + lid + 16];
        c0 = __builtin_amdgcn_wmma_f32_16x16x4_f32(false, a, false, b0,
                                                   (short)0, c0, false, false);
        c1 = __builtin_amdgcn_wmma_f32_16x16x4_f32(false, a, false, b1,
                                                   (short)0, c1, false, false);
    }

    const int mbase = tile * 16 + half * 8;
    if (tile * 16 + 16 <= n) {
        // full tile: no per-row exec-mask churn in the common path
#pragma unroll
        for (int r = 0; r < 8; ++r) {
            xw[(mbase + r) * C_CH + lid]      = c0[r];
            xw[(mbase + r) * C_CH + lid + 16] = c1[r];
        }
    } else {
#pragma unroll
        for (int r = 0; r < 8; ++r) {
            const int row = mbase + r;
            if (row < n) {
                xw[row * C_CH + lid]      = c0[r];
                xw[row * C_CH + lid + 16] = c1[r];
            }
        }
    }
}

// ---------------------------------------------------------------------------
// K4: self-loop contribution: acc[i][c] = xw[i][c] * dinv[i]^2   (fill = 1.0)
// float4-vectorized: one thread handles 4 consecutive channels (b128 ld/st).
// ---------------------------------------------------------------------------
__global__ void k_self(const float* __restrict__ xw, const float* __restrict__ dinv,
                       float* __restrict__ acc, int n) {
    int i = blockIdx.x * blockDim.x + threadIdx.x;          // i indexes float4s
    if (i < n * (C_CH / 4)) {
        const int node = i >> 3;                            // 8 float4s per node
        const float di = dinv[node];
        const float s  = di * di;
        v4f v = ((const v4f*)xw)[i];
        v.x *= s; v.y *= s; v.z *= s; v.w *= s;
        ((v4f*)acc)[i] = v;
    }
}

// ---------------------------------------------------------------------------
// K5: edge aggregation. One wave per edge, lane = channel.
// eidx forced uniform via readfirstlane -> edge scalars become SMEM loads;
// gather of xw[src] and the scatter atomic are coalesced 128B L2 transactions.
// ---------------------------------------------------------------------------
__global__ void k_edges(const int* __restrict__ src, const int* __restrict__ dst,
                        const float* __restrict__ ew, const float* __restrict__ dinv,
                        const float* __restrict__ xw, float* __restrict__ acc, int e) {
    const int lane = threadIdx.x & 31;
    int eidx = blockIdx.x * (blockDim.x >> 5) + (threadIdx.x >> 5);
    eidx = __builtin_amdgcn_readfirstlane(eidx);            // provably wave-uniform
    if (eidx >= e) return;
    const int   s    = src[eidx];
    const int   d    = dst[eidx];
    const float norm = dinv[s] * ew[eidx] * dinv[d];
    const float msg  = xw[s * C_CH + lane] * norm;
    __hip_atomic_fetch_add(&acc[d * C_CH + lane], msg,
                           __ATOMIC_RELAXED, __HIP_MEMORY_SCOPE_AGENT);
}

// ---------------------------------------------------------------------------
// K6: per-channel sum / sumsq partials (bias b cancels under BN -> skipped).
// blockDim must be 256 (8 waves). lane = channel; each wave reads whole rows.
// ---------------------------------------------------------------------------
__global__ void k_stats(const float* __restrict__ acc, float* __restrict__ stats, int n) {
    __shared__ float s_sum[256];
    __shared__ float s_sq[256];
    const int lane = threadIdx.x & 31;
    const int wave = threadIdx.x >> 5;
    float sum = 0.0f, sq = 0.0f;
    int row = blockIdx.x * (blockDim.x >> 5) + wave;
    const int stride = gridDim.x * (blockDim.x >> 5);
    for (; row < n; row += stride) {
        const float v = acc[row * C_CH + lane];
        sum += v;
        sq  += v * v;
    }
    s_sum[threadIdx.x] = sum;
    s_sq[threadIdx.x]  = sq;
    __syncthreads();
    if (wave == 0) {
        float ts = 0.0f, tq = 0.0f;
#pragma unroll
        for (int w = 0; w < 8; ++w) {
            ts += s_sum[w * 32 + lane];
            tq += s_sq[w * 32 + lane];
        }
        __hip_atomic_fetch_add(&stats[lane], ts,
                               __ATOMIC_RELAXED, __HIP_MEMORY_SCOPE_AGENT);
        __hip_atomic_fetch_add(&stats[32 + lane], tq,
                               __ATOMIC_RELAXED, __HIP_MEMORY_SCOPE_AGENT);
    }
}

// ---------------------------------------------------------------------------
// K7: finalize BN: scale = gamma * rsqrt(var+eps), shift = beta - mean*scale
// ---------------------------------------------------------------------------
__global__ void k_bnfin(const float* __restrict__ stats,
                        const float* __restrict__ gamma, const float* __restrict__ beta,
                        float* __restrict__ sc, int n) {
    const int c = threadIdx.x;
    if (c < C_CH) {
        const float invn = 1.0f / (float)n;
        const float mean = stats[c] * invn;
        const float var  = stats[32 + c] * invn - mean * mean;   // biased variance
        const float s    = gamma[c] * rsqrtf(var + BN_EPS);
        sc[c]      = s;
        sc[32 + c] = beta[c] - mean * s;
    }
}

// ---------------------------------------------------------------------------
// K8: in-place BN apply + ReLU on d_out, float4-vectorized (b128 ld/st)
// ---------------------------------------------------------------------------
__global__ void k_apply(float* __restrict__ out, const float* __restrict__ sc,
                        int total4) {
    int i = blockIdx.x * blockDim.x + threadIdx.x;          // i indexes float4s
    if (i < total4) {
        const int c4 = (i & 7) * 4;                         // channel of .x
        v4f v = ((const v4f*)out)[i];
        v.x = v.x * sc[c4 + 0] + sc[32 + c4 + 0];
        v.y = v.y * sc[c4 + 1] + sc[32 + c4 + 1];
        v.z = v.z * sc[c4 + 2] + sc[32 + c4 + 2];
        v.w = v.w * sc[c4 + 3] + sc[32 + c4 + 3];
        v.x = v.x > 0.0f ? v.x : 0.0f;
        v.y = v.y > 0.0f ? v.y : 0.0f;
        v.z = v.z > 0.0f ? v.z : 0.0f;
        v.w = v.w > 0.0f ? v.w : 0.0f;
        ((v4f*)out)[i] = v;
    }
}

// ---------------------------------------------------------------------------
extern "C" void kernel_launch(void* const* d_in, const int* in_sizes, int n_in,
                              void* d_out, int out_size, void* d_ws, size_t ws_size,
                              hipStream_t stream) {
    (void)n_in; (void)out_size; (void)ws_size;
    const float* x     = (const float*)d_in[0];
    const int*   src   = (const int*)d_in[1];
    const int*   dst   = (const int*)d_in[2];
    const float* ew    = (const float*)d_in[3];
    const float* W     = (const float*)d_in[4];
    /* d_in[5] = b : per-channel constant, cancels under BatchNorm -> unused */
    const float* gamma = (const float*)d_in[6];
    const float* beta  = (const float*)d_in[7];

    const int N = in_sizes[0] / C_CH;
    const int E = in_sizes[1];

    float* acc = (float*)d_out;                       // accumulator lives in d_out

    char*  ws  = (char*)d_ws;
    float* xw  = (float*)ws;                          // N * 32 floats
    size_t off = ((size_t)N * C_CH * sizeof(float) + 255) & ~(size_t)255;
    float* dinv = (float*)(ws + off);                 // N floats (deg -> dinv in place)
    off += (((size_t)N * sizeof(float)) + 255) & ~(size_t)255;
    float* stats = (float*)(ws + off);                // 128 floats: sum|sumsq|scale|shift

    const int T = 256;
    const int nVec4 = N * (C_CH / 4);                 // float4 count

    k_init<<<(N + T - 1) / T, T, 0, stream>>>(dinv, stats, N);
    k_deg<<<(E + T - 1) / T, T, 0, stream>>>(dst, ew, dinv, E);
    k_dinv<<<(N + T - 1) / T, T, 0, stream>>>(dinv, N);

    const int ntiles = (N + 15) / 16;                 // 1 wave per 16-row tile
    k_gemm_wmma<<<(ntiles + 7) / 8, 256, 0, stream>>>(x, W, xw, N);

    k_self<<<(nVec4 + T - 1) / T, T, 0, stream>>>(xw, dinv, acc, N);
    k_edges<<<(E + 7) / 8, 256, 0, stream>>>(src, dst, ew, dinv, xw, acc, E);

    k_stats<<<512, 256, 0, stream>>>(acc, stats, N);
    k_bnfin<<<1, 32, 0, stream>>>(stats, gamma, beta, stats + 64, N);
    k_apply<<<(nVec4 + T - 1) / T, T, 0, stream>>>(acc, stats + 64, nVec4);
}